// PlanetoidSAN_54838142435869
// MI455X (gfx1250) — compile-verified
//
#include <hip/hip_runtime.h>
#include <math.h>

typedef __attribute__((ext_vector_type(2))) float v2f;
typedef __attribute__((ext_vector_type(8))) float v8f;
typedef __attribute__((ext_vector_type(4))) unsigned int tdm_v4u;
typedef __attribute__((ext_vector_type(8))) int tdm_v8i;
typedef __attribute__((ext_vector_type(4))) int tdm_v4i;

#define LEAK 0.2f
#define LDS_STRIDE 136   // 128 + 8 pad: the two lane-halves of a B fragment hit disjoint bank groups

static inline unsigned cdivu(size_t a, unsigned b) { return (unsigned)((a + (size_t)b - 1) / b); }

// ---------------- elementwise / fill ----------------

__global__ void fill_kernel(float* __restrict__ p, float v, size_t count) {
    size_t i = (size_t)blockIdx.x * blockDim.x + threadIdx.x;
    size_t stride = (size_t)gridDim.x * blockDim.x;
    for (; i < count; i += stride) p[i] = v;
}

__global__ void binarize_kernel(const float* __restrict__ x, float* __restrict__ y, size_t count4) {
    size_t i = (size_t)blockIdx.x * blockDim.x + threadIdx.x;
    if (i >= count4) return;
    float4 v = ((const float4*)x)[i];
    float4 o = make_float4(v.x != 0.f ? 1.f : 0.f,
                           v.y != 0.f ? 1.f : 0.f,
                           v.z != 0.f ? 1.f : 0.f,
                           v.w != 0.f ? 1.f : 0.f);
    ((float4*)y)[i] = o;
}

// X1f[e] = X0b[i0] * X0b[i1]   (AND of 0/1 features), 32 lanes x float4 per row
__global__ void pair_and_kernel(const float* __restrict__ Xb, const int* __restrict__ idx,
                                float* __restrict__ out, int n) {
    int t = blockIdx.x * blockDim.x + threadIdx.x;
    if (t >= n * 32) return;
    int e = t >> 5, q = (t & 31) * 4;
    const float4 a = *(const float4*)(Xb + (size_t)idx[e * 2 + 0] * 128 + q);
    const float4 b = *(const float4*)(Xb + (size_t)idx[e * 2 + 1] * 128 + q);
    *(float4*)(out + (size_t)e * 128 + q) = make_float4(a.x * b.x, a.y * b.y, a.z * b.z, a.w * b.w);
}

__global__ void triple_and_kernel(const float* __restrict__ Xb, const int* __restrict__ idx,
                                  float* __restrict__ out, int n) {
    int t = blockIdx.x * blockDim.x + threadIdx.x;
    if (t >= n * 32) return;
    int e = t >> 5, q = (t & 31) * 4;
    const float4 a = *(const float4*)(Xb + (size_t)idx[e * 3 + 0] * 128 + q);
    const float4 b = *(const float4*)(Xb + (size_t)idx[e * 3 + 1] * 128 + q);
    const float4 c = *(const float4*)(Xb + (size_t)idx[e * 3 + 2] * 128 + q);
    *(float4*)(out + (size_t)e * 128 + q) =
        make_float4(a.x * b.x * c.x, a.y * b.y * c.y, a.z * b.z * c.z, a.w * b.w * c.w);
}

__global__ void prelu_kernel(float* __restrict__ x, const float* __restrict__ w, size_t count4) {
    size_t i = (size_t)blockIdx.x * blockDim.x + threadIdx.x;
    if (i >= count4) return;
    float wv = *w;
    float4 v = ((float4*)x)[i];
    v.x = v.x >= 0.f ? v.x : wv * v.x;
    v.y = v.y >= 0.f ? v.y : wv * v.y;
    v.z = v.z >= 0.f ? v.z : wv * v.z;
    v.w = v.w >= 0.f ? v.w : wv * v.w;
    ((float4*)x)[i] = v;
}

__global__ void scale_kernel(float* __restrict__ x, float s, size_t count4) {
    size_t i = (size_t)blockIdx.x * blockDim.x + threadIdx.x;
    if (i >= count4) return;
    float4 v = ((float4*)x)[i];
    ((float4*)x)[i] = make_float4(v.x * s, v.y * s, v.z * s, v.w * s);
}

// ---------------- dense GEMM via fp32 WMMA + TDM/async LDS staging ----------------
// H[n x 128] = X[n x 128] @ W[128 x 128] + bias[128]
// Block = 256 threads = 8 waves; block covers 128 rows (8 M-tiles of 16).
// W is staged once per block into LDS with one TENSOR_LOAD_TO_LDS (TDM, TENSORcnt;
// pad_enable produces the bank-padded stride-136 layout natively).
// Each wave then runs a K-loop of 32 x V_WMMA_F32_16X16X4_F32 per N-tile with one
// A-fragment load feeding 8 wmma ops (8 N-tiles / wave).
// A(16x4 f32): lanes 0-15 M=0..15 {K=kb,kb+1}, lanes 16-31 M=0..15 {K=kb+2,kb+3}
// B(4x16 f32): lanes hold N=lane&15; half selects K pair (mirrors A layout)
// C/D(16x16 f32): VGPR i -> M = i + (lane>=16)*8, N = lane&15
__global__ void __launch_bounds__(256)
gemm128_wmma(const float* __restrict__ X, const float* __restrict__ W,
             const float* __restrict__ bias, float* __restrict__ H, int n) {
    __shared__ float sW[128 * LDS_STRIDE];
    const int t = threadIdx.x;

#if __has_builtin(__builtin_amdgcn_tensor_load_to_lds) && \
    __has_builtin(__builtin_amdgcn_s_wait_tensorcnt)
    // ---- TDM: one tensor DMA per block stages the whole 128x128 W tile ----
    if (t < 32) {  // one wave issues; TDM ignores EXEC, tracked by TENSORcnt
        unsigned long long g = (unsigned long long)(uintptr_t)W;
        unsigned lds = (unsigned)(uintptr_t)sW;
        // D# group0: count=1 | lds_addr | global_addr[56:0] | type=2
        tdm_v4u g0 = {1u, lds, (unsigned)g,
                      (unsigned)((g >> 32) & 0x01FFFFFFull) | (2u << 30)};
        // D# group1: data_size=4B, pad every 128 DWORDs by 8 DWORDs (-> stride 136),
        // tensor_dim0=128, tensor_dim1=128, tile_dim0=128, tile_dim1=128, dim0_stride=128
        tdm_v8i g1 = {(int)0x0F920000,      // data_size=2<<16 | pad_en<<20 | intv=6<<22 | amt=7<<25
                      (int)(128u << 16),    // tensor_dim0[15:0] in bits 63:48
                      (int)(128u << 16),    // tensor_dim1[15:0] in bits 95:80
                      (int)(128u << 16),    // tile_dim0 in bits 127:112
                      128,                  // tile_dim1
                      128,                  // tensor_dim0_stride[31:0]
                      0, 0};
        tdm_v4i g2 = {0, 0, 0, 0};
        tdm_v4i g3 = {0, 0, 0, 0};
        tdm_v8i g4 = {0, 0, 0, 0, 0, 0, 0, 0};  // clang-23 6-arg form: extra group, zero-filled
        __builtin_amdgcn_tensor_load_to_lds(g0, g1, g2, g3, g4, 0);
        __builtin_amdgcn_s_wait_tensorcnt(0);
    }
    __syncthreads();
#else
    // ---- fallback: per-lane async global->LDS copies (16 x b128 per thread) ----
#pragma unroll
    for (int i = 0; i < 16; ++i) {
        int c   = i * 256 + t;        // float4 chunk id, 0..4095
        int row = c >> 5;             // 32 chunks per 128-float row
        int q4  = (c & 31) * 4;       // float index within row
        unsigned lds = (unsigned)(uintptr_t)(sW + row * LDS_STRIDE + q4);
        unsigned long long g = (unsigned long long)(uintptr_t)(W + (size_t)row * 128 + q4);
        asm volatile("global_load_async_to_lds_b128 %0, %1, off" ::"v"(lds), "v"(g) : "memory");
    }
    asm volatile("s_wait_asynccnt 0x0" ::: "memory");
    __syncthreads();
#endif

    const int wave  = t >> 5;
    const int lane  = t & 31;
    const int tileM = blockIdx.x * 8 + wave;
    const int half  = lane >> 4;  // 0 or 1
    const int l15   = lane & 15;

    if (tileM * 16 >= n) return;  // inactive waves already did staging + barrier

    const float* Xrow = X + (size_t)(tileM * 16 + l15) * 128;
    v8f acc[8] = {};
    for (int k = 0; k < 128; k += 4) {
        int ka = k + half * 2;
        v2f a;
        a.x = Xrow[ka];
        a.y = Xrow[ka + 1];
        const float* wrow0 = sW + ka * LDS_STRIDE + l15;
        const float* wrow1 = wrow0 + LDS_STRIDE;
#pragma unroll
        for (int tn = 0; tn < 8; ++tn) {
            v2f b;
            b.x = wrow0[tn * 16];
            b.y = wrow1[tn * 16];
            acc[tn] = __builtin_amdgcn_wmma_f32_16x16x4_f32(false, a, false, b, (short)0,
                                                            acc[tn], false, false);
        }
    }
    float* outbase = H + (size_t)(tileM * 16 + half * 8) * 128;
#pragma unroll
    for (int tn = 0; tn < 8; ++tn) {
        int colN = tn * 16 + l15;
        float bv = bias[colN];
#pragma unroll
        for (int i = 0; i < 8; ++i) outbase[(size_t)i * 128 + colN] = acc[tn][i] + bv;
    }
}

// ---------------- GAT building blocks ----------------

// per-row dot products: s_src[i] = <h[i], a_src>, s_dst[i] = <h[i], a_dst>
__global__ void row_dots_kernel(const float* __restrict__ h, const float* __restrict__ a_src,
                                const float* __restrict__ a_dst, float* __restrict__ s_src,
                                float* __restrict__ s_dst, int n) {
    int t = blockIdx.x * blockDim.x + threadIdx.x;
    int row = t >> 5, lane = t & 31;
    if (row >= n) return;
    float4 v  = *(const float4*)(h + (size_t)row * 128 + lane * 4);
    float4 as = ((const float4*)a_src)[lane];
    float4 ad = ((const float4*)a_dst)[lane];
    float ss = v.x * as.x + v.y * as.y + v.z * as.z + v.w * as.w;
    float sd = v.x * ad.x + v.y * ad.y + v.z * ad.z + v.w * ad.w;
#pragma unroll
    for (int off = 16; off > 0; off >>= 1) {
        ss += __shfl_down(ss, off, 32);
        sd += __shfl_down(sd, off, 32);
    }
    if (lane == 0) { s_src[row] = ss; s_dst[row] = sd; }
}

__device__ inline void atomicMaxFloat(float* addr, float val) {
    if (val >= 0.f) atomicMax((int*)addr, __float_as_int(val));
    else            atomicMin((unsigned int*)addr, __float_as_uint(val));
}

// e = leaky_relu(s_src[r] + s_dst[c]); segment max into m[r]
__global__ void edge_logits_kernel(const int* __restrict__ rows, const int* __restrict__ cols,
                                   const float* __restrict__ s_src, const float* __restrict__ s_dst,
                                   float* __restrict__ escr, float* __restrict__ m, int nnz) {
    int e = blockIdx.x * blockDim.x + threadIdx.x;
    if (e >= nnz) return;
    int r = rows[e], c = cols[e];
    float v = s_src[r] + s_dst[c];
    v = v >= 0.f ? v : LEAK * v;
    escr[e] = v;
    atomicMaxFloat(&m[r], v);
}

// ex = exp(e - m[r]); segment sum into den[r]
__global__ void edge_exp_kernel(const int* __restrict__ rows, float* __restrict__ escr,
                                const float* __restrict__ m, float* __restrict__ den, int nnz) {
    int e = blockIdx.x * blockDim.x + threadIdx.x;
    if (e >= nnz) return;
    int r = rows[e];
    float ex = expf(escr[e] - m[r]);
    escr[e] = ex;
    atomicAdd(&den[r], ex);
}

// out[r] += (ex[e]/(den[r]+eps)) * h[c]   (one wave per edge, float4 per lane)
__global__ void gat_scatter_kernel(const int* __restrict__ rows, const int* __restrict__ cols,
                                   const float* __restrict__ escr, const float* __restrict__ den,
                                   const float* __restrict__ h, float* __restrict__ out, int nnz) {
    int t = blockIdx.x * blockDim.x + threadIdx.x;
    int e = t >> 5;
    if (e >= nnz) return;
    int lane = t & 31;
    int r = rows[e], c = cols[e];
    float coeff = escr[e] / (den[r] + 1e-16f);
    float4 v = *(const float4*)(h + (size_t)c * 128 + lane * 4);
    float* o = out + (size_t)r * 128 + lane * 4;
    atomicAdd(o + 0, coeff * v.x);
    atomicAdd(o + 1, coeff * v.y);
    atomicAdd(o + 2, coeff * v.z);
    atomicAdd(o + 3, coeff * v.w);
}

// out[r] += vals[e] * X[c]
__global__ void spmm_scatter_kernel(const int* __restrict__ rows, const int* __restrict__ cols,
                                    const float* __restrict__ vals, const float* __restrict__ X,
                                    float* __restrict__ out, int nnz) {
    int t = blockIdx.x * blockDim.x + threadIdx.x;
    int e = t >> 5;
    if (e >= nnz) return;
    int lane = t & 31;
    int r = rows[e], c = cols[e];
    float v = vals[e];
    float4 x = *(const float4*)(X + (size_t)c * 128 + lane * 4);
    float* o = out + (size_t)r * 128 + lane * 4;
    atomicAdd(o + 0, v * x.x);
    atomicAdd(o + 1, v * x.y);
    atomicAdd(o + 2, v * x.z);
    atomicAdd(o + 3, v * x.w);
}

// ---------------- host-side helpers ----------------

static void run_gemm(const float* X, const float* W, const float* b, float* H, int n,
                     hipStream_t s) {
    // 8 M-tiles (128 rows) per block of 256 threads
    gemm128_wmma<<<cdivu((size_t)(n / 16), 8), 256, 0, s>>>(X, W, b, H, n);
}

static void run_gat(const float* h, const int* rows, const int* cols, int nnz,
                    const float* a_src, const float* a_dst, int n,
                    float* ssrc, float* sdst, float* mbuf, float* den, float* escr,
                    float* out, hipStream_t s) {
    row_dots_kernel<<<cdivu((size_t)n * 32, 256), 256, 0, s>>>(h, a_src, a_dst, ssrc, sdst, n);
    fill_kernel<<<cdivu(n, 256), 256, 0, s>>>(mbuf, -INFINITY, n);
    fill_kernel<<<cdivu(n, 256), 256, 0, s>>>(den, 0.f, n);
    edge_logits_kernel<<<cdivu(nnz, 256), 256, 0, s>>>(rows, cols, ssrc, sdst, escr, mbuf, nnz);
    edge_exp_kernel<<<cdivu(nnz, 256), 256, 0, s>>>(rows, escr, mbuf, den, nnz);
    gat_scatter_kernel<<<cdivu((size_t)nnz * 32, 256), 256, 0, s>>>(rows, cols, escr, den, h, out,
                                                                    nnz);
}

static void run_spmm(const int* rows, const int* cols, const float* vals, const float* X,
                     float* out, int nnz, hipStream_t s) {
    spmm_scatter_kernel<<<cdivu((size_t)nnz * 32, 256), 256, 0, s>>>(rows, cols, vals, X, out, nnz);
}

extern "C" void kernel_launch(void* const* d_in, const int* in_sizes, int n_in,
                              void* d_out, int out_size, void* d_ws, size_t ws_size,
                              hipStream_t stream) {
    (void)n_in; (void)ws_size; (void)out_size;
    // ---- raw inputs (setup_inputs dict order) ----
    const float* X0     = (const float*)d_in[0];
    const int*   X1_idx = (const int*)d_in[1];
    const int*   X2_idx = (const int*)d_in[2];
    const int n0 = in_sizes[0] / 128;
    const int n1 = in_sizes[1] / 2;
    const int n2 = in_sizes[2] / 3;

    const int *L0r = (const int*)d_in[3],  *L0c = (const int*)d_in[4];
    const float* L0v = (const float*)d_in[5];   int nnzL0  = in_sizes[5];
    const int *L1r = (const int*)d_in[6],  *L1c = (const int*)d_in[7];
    const float* L1v = (const float*)d_in[8];   int nnzL1  = in_sizes[8];
    const int *L1ur = (const int*)d_in[9], *L1uc = (const int*)d_in[10];
    int nnzL1u = in_sizes[11];
    const int *L1dr = (const int*)d_in[12], *L1dc = (const int*)d_in[13];
    int nnzL1d = in_sizes[14];
    const int *L2r = (const int*)d_in[15], *L2c = (const int*)d_in[16];
    const float* L2v = (const float*)d_in[17];  int nnzL2  = in_sizes[17];
    const int *B1r = (const int*)d_in[18], *B1c = (const int*)d_in[19];
    const float* B1v = (const float*)d_in[20];  int nnzB1  = in_sizes[20];
    const int *B2r = (const int*)d_in[21], *B2c = (const int*)d_in[22];
    const float* B2v = (const float*)d_in[23];  int nnzB2  = in_sizes[23];

    // ---- params: jax tree-flatten order (sorted dict keys) ----
    const float* e_d_W    = (const float*)d_in[24];
    const float* e_d_adst = (const float*)d_in[25];
    const float* e_d_asrc = (const float*)d_in[26];
    const float* e_d_b    = (const float*)d_in[27];
    const float* e_pW     = (const float*)d_in[28];
    const float* e_pb     = (const float*)d_in[29];
    const float* e_u_W    = (const float*)d_in[30];
    const float* e_u_adst = (const float*)d_in[31];
    const float* e_u_asrc = (const float*)d_in[32];
    const float* e_u_b    = (const float*)d_in[33];
    const float* n_d_W    = (const float*)d_in[34];
    const float* n_d_adst = (const float*)d_in[35];
    const float* n_d_asrc = (const float*)d_in[36];
    const float* n_d_b    = (const float*)d_in[37];
    const float* n_pW     = (const float*)d_in[38];
    const float* n_pb     = (const float*)d_in[39];
    const float* prelu_w  = (const float*)d_in[40];
    const float* t_pW     = (const float*)d_in[41];
    const float* t_pb     = (const float*)d_in[42];
    const float* t_u_W    = (const float*)d_in[43];
    const float* t_u_adst = (const float*)d_in[44];
    const float* t_u_asrc = (const float*)d_in[45];
    const float* t_u_b    = (const float*)d_in[46];
    const float* tri_W    = (const float*)d_in[47];
    const float* tri_b    = (const float*)d_in[48];

    // ---- workspace layout (floats) ----
    float* ws = (float*)d_ws;
    size_t off = 0;
    float* X0b  = ws + off; off += (size_t)n0 * 128;
    float* X1f  = ws + off; off += (size_t)n1 * 128;   // later reused as T1 = B2 @ X2o
    float* X2f  = ws + off; off += (size_t)n2 * 128;
    float* G1   = ws + off; off += (size_t)n1 * 128;   // shared GEMM output scratch
    float* ACC1 = ws + off; off += (size_t)n1 * 128;   // X1o
    float* ACC2 = ws + off; off += (size_t)n2 * 128;   // X2o
    float* ssrc = ws + off; off += (size_t)n1;
    float* sdst = ws + off; off += (size_t)n1;
    float* mbuf = ws + off; off += (size_t)n1;
    float* den  = ws + off; off += (size_t)n1;
    float* escr = ws + off; off += (size_t)(nnzL1u > nnzL0 ? nnzL1u : nnzL0);

    float* OUT = (float*)d_out;   // doubles as the node-level accumulator (ACC0)

    const unsigned B = 256;

    // ---- phase A: features ----
    binarize_kernel<<<cdivu((size_t)n0 * 32, B), B, 0, stream>>>(X0, X0b, (size_t)n0 * 32);
    pair_and_kernel<<<cdivu((size_t)n1 * 32, B), B, 0, stream>>>(X0b, X1_idx, X1f, n1);
    triple_and_kernel<<<cdivu((size_t)n2 * 32, B), B, 0, stream>>>(X0b, X2_idx, X2f, n2);

    // ---- layer n (nodes): GAT(L0) + SpMM(L0, X0b@pW+pb), PReLU ----
    fill_kernel<<<cdivu((size_t)n0 * 128, B), B, 0, stream>>>(OUT, 0.f, (size_t)n0 * 128);
    run_gemm(X0b, n_d_W, n_d_b, G1, n0, stream);
    run_gat(G1, L0r, L0c, nnzL0, n_d_asrc, n_d_adst, n0, ssrc, sdst, mbuf, den, escr, OUT, stream);
    run_gemm(X0b, n_pW, n_pb, G1, n0, stream);
    run_spmm(L0r, L0c, L0v, G1, OUT, nnzL0, stream);
    prelu_kernel<<<cdivu((size_t)n0 * 32, B), B, 0, stream>>>(OUT, prelu_w, (size_t)n0 * 32);

    // ---- layer e (edges): GAT(L1u) + GAT(L1d) + SpMM(L1), PReLU ----
    fill_kernel<<<cdivu((size_t)n1 * 128, B), B, 0, stream>>>(ACC1, 0.f, (size_t)n1 * 128);
    run_gemm(X1f, e_u_W, e_u_b, G1, n1, stream);
    run_gat(G1, L1ur, L1uc, nnzL1u, e_u_asrc, e_u_adst, n1, ssrc, sdst, mbuf, den, escr, ACC1,
            stream);
    run_gemm(X1f, e_d_W, e_d_b, G1, n1, stream);
    run_gat(G1, L1dr, L1dc, nnzL1d, e_d_asrc, e_d_adst, n1, ssrc, sdst, mbuf, den, escr, ACC1,
            stream);
    run_gemm(X1f, e_pW, e_pb, G1, n1, stream);
    run_spmm(L1r, L1c, L1v, G1, ACC1, nnzL1, stream);
    prelu_kernel<<<cdivu((size_t)n1 * 32, B), B, 0, stream>>>(ACC1, prelu_w, (size_t)n1 * 32);

    // ---- layer t (triangles): GAT(L2) + SpMM(L2), PReLU ----
    fill_kernel<<<cdivu((size_t)n2 * 128, B), B, 0, stream>>>(ACC2, 0.f, (size_t)n2 * 128);
    run_gemm(X2f, t_u_W, t_u_b, G1, n2, stream);
    run_gat(G1, L2r, L2c, nnzL2, t_u_asrc, t_u_adst, n2, ssrc, sdst, mbuf, den, escr, ACC2, stream);
    run_gemm(X2f, t_pW, t_pb, G1, n2, stream);
    run_spmm(L2r, L2c, L2v, G1, ACC2, nnzL2, stream);
    prelu_kernel<<<cdivu((size_t)n2 * 32, B), B, 0, stream>>>(ACC2, prelu_w, (size_t)n2 * 32);

    // ---- tri = (B2 @ X2o) @ tri_W + tri_b ----
    float* T1 = X1f;  // X1f is dead now
    fill_kernel<<<cdivu((size_t)n1 * 128, B), B, 0, stream>>>(T1, 0.f, (size_t)n1 * 128);
    run_spmm(B2r, B2c, B2v, ACC2, T1, nnzB2, stream);
    run_gemm(T1, tri_W, tri_b, G1, n1, stream);   // G1 = tri

    // ---- out = (X0o + B1@X1o + B1@tri) / 3 ----
    run_spmm(B1r, B1c, B1v, ACC1, OUT, nnzB1, stream);
    run_spmm(B1r, B1c, B1v, G1, OUT, nnzB1, stream);
    scale_kernel<<<cdivu((size_t)n0 * 32, B), B, 0, stream>>>(OUT, 1.f / 3.f, (size_t)n0 * 32);
}